// StabilizedButterflyLayer_35845797052976
// MI455X (gfx1250) — compile-verified
//
#include <hip/hip_runtime.h>
#include <stdint.h>

typedef __attribute__((ext_vector_type(2))) float v2f;
typedef __attribute__((ext_vector_type(8))) float v8f;

#define N_ELEM   4096
#define N_PAIRS  2048
#define BATCH_T  16          // batch rows per workgroup / per WMMA tile
#define ROWW     4100        // LDS words per batch row (4096 + 4 pad, keeps 16B align)

// ---------------------------------------------------------------------------
// Prep: build 3 rounds x 256 matrices of 16x16 (product of 4 rotation stages)
// mats[r][c][row][col], row-major 16x16 per matrix. One thread per column.
// ---------------------------------------------------------------------------
__global__ void butterfly_prep(const float* __restrict__ angles,
                               float* __restrict__ mats)
{
    int id = blockIdx.x * blockDim.x + threadIdx.x;   // 3*256*16 = 12288
    if (id >= 3 * 256 * 16) return;
    int k = id & 15;          // column of the matrix
    int c = (id >> 4) & 255;  // tile-column index within the round
    int r = id >> 12;         // round 0..2

    float v[16];
#pragma unroll
    for (int a = 0; a < 16; ++a) v[a] = (a == k) ? 1.0f : 0.0f;

#pragma unroll
    for (int sp = 0; sp < 4; ++sp) {
        int s = r * 4 + sp;
        const float* ang = angles + s * N_PAIRS;
        int d = 1 << sp;
#pragma unroll
        for (int a0 = 0; a0 < 16; ++a0) {
            if (a0 & d) continue;                      // a0 has bit sp clear
            int a1 = a0 + d;
            int pil = ((a0 >> (sp + 1)) << sp) | (a0 & (d - 1)); // 0..7
            int p;
            if (r == 0)      p = (c << 3) + pil;                            // 8q+pil
            else if (r == 1) p = ((((c >> 4) << 3) + pil) << 4) + (c & 15); // 16*(8Q+pil)+j
            else             p = (pil << 8) + c;                            // 256*pil+j
            float th = ang[p];
            float sn, cs;
            __sincosf(th, &sn, &cs);
            float t0 = v[a0], t1 = v[a1];
            v[a0] = cs * t0 - sn * t1;
            v[a1] = sn * t0 + cs * t1;
        }
    }

    float* M = mats + ((size_t)(id >> 4) << 8);       // (r*256+c)*256
#pragma unroll
    for (int a = 0; a < 16; ++a) M[a * 16 + k] = v[a];
}

// ---------------------------------------------------------------------------
// Main: one workgroup = 16 batch rows x 4096 elements, fully in LDS
// (layout lds[batch][elem], row stride ROWW words).
// 3 rounds; each round = 256 tile GEMMs D = M(16x16) * X(16a x 16batch)
// via 4x V_WMMA_F32_16X16X4_F32. 256 threads = 8 waves, 32 tiles/wave/round.
// Global<->LDS movement uses the CDNA5 async-to/from-LDS engine (ASYNCcnt).
// ---------------------------------------------------------------------------
__global__ void butterfly_main(const float* __restrict__ x,
                               const float* __restrict__ mats,
                               float* __restrict__ out)
{
    extern __shared__ float lds[];                    // BATCH_T * ROWW floats

    const int wg   = blockIdx.x;                      // 256 workgroups
    const int tid  = threadIdx.x;                     // 256 threads
    const int lane = tid & 31;
    const int wave = tid >> 5;                        // 0..7
    const int hi   = lane >> 4;                       // 0 / 1 (lane half)
    const int lo   = lane & 15;

    // ---- async load: 16 contiguous rows straight into LDS, one row per batch ----
#pragma unroll
    for (int rb = 0; rb < 2; ++rb) {
        int b16 = wave * 2 + rb;
        const float* src = x + (size_t)(wg * BATCH_T + b16) * N_ELEM;
#pragma unroll 1
        for (int it = 0; it < 32; ++it) {
            int e0 = it * 128 + lane * 4;
            uint32_t la = (uint32_t)(uintptr_t)(&lds[b16 * ROWW + e0]);
            const float* ga = src + e0;
            asm volatile("global_load_async_to_lds_b128 %0, %1, off"
                         :: "v"(la), "v"(ga) : "memory");
        }
    }
    asm volatile("s_wait_asynccnt 0x0" ::: "memory");
    __syncthreads();

    // ---- 3 rounds of 256 tile matmuls, A-operand pipelined one tile ahead ----
    for (int r = 0; r < 3; ++r) {
        const float* Mr = mats + ((size_t)r << 16);   // r*256 matrices
        const int c0 = wave * 32;

        v2f a_nxt[4];
        {
            const float* M = Mr + ((size_t)c0 << 8);
#pragma unroll
            for (int q = 0; q < 4; ++q)
                a_nxt[q] = *(const v2f*)(M + lo * 16 + q * 4 + (hi << 1));
        }

#pragma unroll 1
        for (int t = 0; t < 32; ++t) {
            int c = c0 + t;
            int ebase, estride;
            if (r == 0)      { ebase = c << 4;                     estride = 1;   }
            else if (r == 1) { ebase = ((c >> 4) << 8) + (c & 15); estride = 16;  }
            else             { ebase = c;                          estride = 256; }

            v2f aops[4] = { a_nxt[0], a_nxt[1], a_nxt[2], a_nxt[3] };
            if (t < 31) {                              // prefetch next tile's matrix
                const float* M = Mr + ((size_t)(c + 1) << 8);
#pragma unroll
                for (int q = 0; q < 4; ++q)
                    a_nxt[q] = *(const v2f*)(M + lo * 16 + q * 4 + (hi << 1));
            }

            const int base = lo * ROWW + ebase;        // lo = batch lane (B/D operand)
            v8f acc = {};
#pragma unroll
            for (int q = 0; q < 4; ++q) {
                int kq = q * 4 + (hi << 1);            // K pair base for this lane half
                v2f b;
                b.x = lds[base + (kq + 0) * estride];
                b.y = lds[base + (kq + 1) * estride];
                acc = __builtin_amdgcn_wmma_f32_16x16x4_f32(
                          false, aops[q], false, b, (short)0, acc, false, false);
            }

            // D write-back (in place; tiles are element-disjoint within a round)
#pragma unroll
            for (int rr = 0; rr < 8; ++rr)
                lds[base + (rr + (hi << 3)) * estride] = acc[rr];
        }
        __syncthreads();
    }

    // ---- async store: the per-batch rows in LDS straight back to global ----
#pragma unroll
    for (int rb = 0; rb < 2; ++rb) {
        int b16 = wave * 2 + rb;
        const float* dst = out + (size_t)(wg * BATCH_T + b16) * N_ELEM;
#pragma unroll 1
        for (int it = 0; it < 32; ++it) {
            int e0 = it * 128 + lane * 4;
            uint32_t la = (uint32_t)(uintptr_t)(&lds[b16 * ROWW + e0]);
            const float* ga = dst + e0;
            asm volatile("global_store_async_from_lds_b128 %0, %1, off"
                         :: "v"(ga), "v"(la) : "memory");
        }
    }
    asm volatile("s_wait_asynccnt 0x0" ::: "memory");
}

// ---------------------------------------------------------------------------
extern "C" void kernel_launch(void* const* d_in, const int* in_sizes, int n_in,
                              void* d_out, int out_size, void* d_ws, size_t ws_size,
                              hipStream_t stream)
{
    (void)in_sizes; (void)n_in; (void)out_size; (void)ws_size;
    const float* x      = (const float*)d_in[0];   // [4096, 4096] f32
    const float* angles = (const float*)d_in[1];   // [12, 2048]   f32
    float*       out    = (float*)d_out;           // [4096, 4096] f32
    float*       mats   = (float*)d_ws;            // 3*256*256 floats = 768 KB

    butterfly_prep<<<48, 256, 0, stream>>>(angles, mats);

    const size_t ldsBytes = (size_t)BATCH_T * ROWW * sizeof(float); // 256.25 KB
    butterfly_main<<<4096 / BATCH_T, 256, ldsBytes, stream>>>(x, mats, out);
}